// GAT_48653389529416
// MI455X (gfx1250) — compile-verified
//
#include <hip/hip_runtime.h>
#include <hip/hip_bf16.h>
#include <math.h>

typedef __attribute__((ext_vector_type(2))) float v2f;
typedef __attribute__((ext_vector_type(8))) float v8f;

// ---------------- problem constants ----------------
constexpr int PASTN  = 256;
constexpr int FUT    = 64;
constexpr int SB     = 64;
constexpr int NT     = 320;
constexpr int DM     = 106;
constexpr int OD     = 35;            // feature dim
constexpr int OD1    = 34;            // O-1
constexpr int ODP    = 36;            // padded 35 -> 36 (K pad)
constexpr int ODP48  = 48;            // padded 34 -> 48 (N pad)
constexpr int R1     = SB * PASTN;    // 16384
constexpr int R2     = SB * FUT;      // 4096
constexpr int RT     = R1 + R2;       // 20480

// ---------------- workspace layout (floats) ----------------
constexpr size_t oA    = 0;                     // 320*320
constexpr size_t oAhP  = oA    + 102400;        // 256*256
constexpr size_t oDgP  = oAhP  + 65536;         // 256
constexpr size_t oAhF  = oDgP  + 256;           // 64*64
constexpr size_t oDgF  = oAhF  + 4096;          // 64
constexpr size_t oAt   = oDgF  + 64;            // 64*256
constexpr size_t oQ    = oAt   + 16384;         // 20480
constexpr size_t oYh   = oQ    + 20480;         // 4096
constexpr size_t oT7   = oYh   + 4096;          // 4096
constexpr size_t oSq   = oT7   + 4096;          // 20480
constexpr size_t oWp1  = oSq   + 20480;         // 36*128
constexpr size_t oWp3  = oWp1  + 4608;          // 128*48
constexpr size_t oG1W1 = oWp3  + 6144;          // 36*64
constexpr size_t oG1W2 = oG1W1 + 2304;          // 64*48
constexpr size_t oG2W1 = oG1W2 + 3072;          // 36*64
constexpr size_t oTh   = oG2W1 + 2304;          // 36*48
constexpr size_t oP1   = oTh   + 1728;          // 20480*128 (H1; later t1,t2; later XC)
constexpr size_t oP2   = oP1   + 2621440;       // 20480*128 (H2; later t3,x1g; later t5,t6)
constexpr size_t oF    = oP2   + 2621440;       // 20480*36  (F; later XK)
constexpr size_t oH3   = oF    + 737280;        // 20480*48
constexpr size_t oX0   = oH3   + 983040;        // 20480*36
constexpr size_t oZ    = oX0   + 737280;        // 20480*36
constexpr size_t oX2B  = oZ    + 737280;        // 4096*36

// aliased sub-buffers
constexpr size_t oT1  = oP1;                    // 16384*64
constexpr size_t oT2  = oP1 + 1048576;          // 16384*64
constexpr size_t oXC  = oP1;                    // 20480*36 (after t1/t2 dead)
constexpr size_t oT3  = oP2;                    // 16384*48
constexpr size_t oX1G = oP2 + 786432;           // 16384*48
constexpr size_t oT5  = oP2;                    // 4096*64
constexpr size_t oT6  = oP2 + 262144;           // 4096*64
constexpr size_t oXK  = oF;                     // 20480*36 (after F dead)

// ---------------- reductions ----------------
__device__ inline float blk_max(float v, float* red) {
    int t = threadIdx.x;
    red[t] = v; __syncthreads();
    for (int s = blockDim.x >> 1; s > 0; s >>= 1) {
        if (t < s) red[t] = fmaxf(red[t], red[t + s]);
        __syncthreads();
    }
    float r = red[0]; __syncthreads(); return r;
}
__device__ inline float blk_sum(float v, float* red) {
    int t = threadIdx.x;
    red[t] = v; __syncthreads();
    for (int s = blockDim.x >> 1; s > 0; s >>= 1) {
        if (t < s) red[t] += red[t + s];
        __syncthreads();
    }
    float r = red[0]; __syncthreads(); return r;
}

// ---------------- WMMA fp32 16x16x4 inner loop (guard-free) ----------------
// A lanes: row = lane&15, k = k0 + 2*(lane>>4) + {0,1}  (contiguous b64)
// B lanes: col = lane&15, same k pairing; B staged transposed in LDS (ds b64)
// C/D:     m = vgpr + 8*(lane>>4), n = lane&15
template <typename PA, typename PB>
__device__ inline v8f wmma_loop(PA Arow, PB Bcol, int Kp, int koff) {
    v8f c = {0.f,0.f,0.f,0.f,0.f,0.f,0.f,0.f};
#pragma unroll 4
    for (int k0 = 0; k0 < Kp; k0 += 4) {
        int ka = k0 + koff;
        v2f a, b;
        a.x = Arow[ka]; a.y = Arow[ka + 1];
        b.x = Bcol[ka]; b.y = Bcol[ka + 1];
        c = __builtin_amdgcn_wmma_f32_16x16x4_f32(false, a, false, b,
                                                  (short)0, c, false, false);
    }
    return c;
}

// ================= kernels =================

// A = row-softmax over tril of relu(M M^T)
__global__ void k_rowA(const float* __restrict__ Mg, float* __restrict__ Aws,
                       float* __restrict__ Aout) {
    __shared__ float mrow[DM];
    __shared__ float red[256];
    const int i = blockIdx.x, t = threadIdx.x;
    for (int k = t; k < DM; k += 256) mrow[k] = Mg[(size_t)i * DM + k];
    __syncthreads();
    float sv[2]; int jv[2]; int nv = 0;
    for (int j = t; j < NT; j += 256) {
        if (j <= i) {
            float d = 0.f;
            const float* mj = Mg + (size_t)j * DM;
            for (int k = 0; k < DM; ++k) d += mrow[k] * mj[k];
            sv[nv] = fmaxf(d, 0.f); jv[nv] = j; ++nv;
        }
    }
    float lmax = -__builtin_inff();
    for (int u = 0; u < nv; ++u) lmax = fmaxf(lmax, sv[u]);
    float mx = blk_max(lmax, red);
    float lsum = 0.f, ev[2];
    for (int u = 0; u < nv; ++u) { ev[u] = __expf(sv[u] - mx); lsum += ev[u]; }
    float sm = blk_sum(lsum, red);
    for (int u = 0; u < nv; ++u) {
        float v = ev[u] / sm;
        Aws[(size_t)i * NT + jv[u]]  = v;
        Aout[(size_t)i * NT + jv[u]] = v;
    }
    for (int j = t; j < NT; j += 256)
        if (j > i) { Aws[(size_t)i * NT + j] = 0.f; Aout[(size_t)i * NT + j] = 0.f; }
}

// features, padded stride 36 (col 35 = 0)
__global__ void k_feat(const float* __restrict__ eh, const float* __restrict__ ed,
                       const float* __restrict__ em,
                       const int* __restrict__ c1, const float* __restrict__ f1,
                       const int* __restrict__ c2, const float* __restrict__ f2,
                       float* __restrict__ F) {
    int id = blockIdx.x * blockDim.x + threadIdx.x;
    if (id >= RT * ODP) return;
    int r = id / ODP, c = id % ODP;
    const int*   cat  = (r < R1) ? (c1 + (size_t)r * 3) : (c2 + (size_t)(r - R1) * 3);
    const float* cont = (r < R1) ? (f1 + (size_t)r * 3) : (f2 + (size_t)(r - R1) * 3);
    float v;
    if      (c < 16) v = eh[cat[0] * 16 + c];
    else if (c < 24) v = ed[cat[1] * 8 + (c - 16)];
    else if (c < 32) v = em[cat[2] * 8 + (c - 24)];
    else if (c < OD) v = cont[c - 32];
    else             v = 0.f;
    F[id] = v;
}

// zero-padded weight copy: dst(Kp x Np) from src(K x N)
__global__ void k_padw(const float* __restrict__ src, int K, int N,
                       float* __restrict__ dst, int Kp, int Np) {
    int id = blockIdx.x * blockDim.x + threadIdx.x;
    if (id >= Kp * Np) return;
    int k = id / Np, n = id % Np;
    dst[id] = (k < K && n < N) ? src[k * N + n] : 0.f;
}

// Generic (optionally batched) WMMA GEMM: C = act(A@B + bias)
// Block = 128 thr = 4 waves -> 64x16 C macro-tile. B tile staged transposed in LDS.
// Requires: M % 64 == 0, Kp % 4 == 0, all operand buffers padded so loads need no guards.
__global__ void k_gemm(const float* __restrict__ A, int lda, size_t strA,
                       const float* __restrict__ B, int ldb, size_t strB,
                       const float* __restrict__ bias,
                       float* __restrict__ C, int ldc, size_t strC,
                       int M, int N, int Kp, int act) {
    __shared__ float bs[16 * 256];
    const int bt = blockIdx.y;
    A += (size_t)bt * strA; B += (size_t)bt * strB; C += (size_t)bt * strC;
    const int ntn = (N + 15) >> 4;
    const int bm = blockIdx.x / ntn, bn = blockIdx.x % ntn;
    const int n0 = bn << 4;
    for (int idx = threadIdx.x; idx < (Kp << 4); idx += 128) {
        int cc = idx / Kp, k = idx - cc * Kp;
        bs[idx] = B[(size_t)k * ldb + n0 + cc];
    }
    __syncthreads();
    const int lane = threadIdx.x & 31;
    const int r = lane & 15, koff = (lane >> 4) << 1;
    const int m0 = (bm << 6) + ((threadIdx.x >> 5) << 4);
    const float* Arow = A + (size_t)(m0 + r) * lda;
    __builtin_prefetch(Arow, 0, 3);
    v8f acc = wmma_loop(Arow, bs + r * Kp, Kp, koff);
    const int col = lane & 15, rhi = (lane >> 4) << 3;
    for (int i = 0; i < 8; ++i) {
        int m = m0 + rhi + i, n = n0 + col;
        if (n < N) {
            float v = acc[i] + (bias ? bias[n] : 0.f);
            if (act) v = fmaxf(v, 0.f);
            C[(size_t)m * ldc + n] = v;
        } else if (n < ldc) {
            C[(size_t)m * ldc + n] = 0.f;   // zero-fill pad columns
        }
    }
}

// x0full stride 36: [0..33]=H3(48), [34]=cont z, [35]=0
__global__ void k_x0build(const float* __restrict__ H3,
                          const float* __restrict__ f1, const float* __restrict__ f2,
                          float* __restrict__ X0) {
    int id = blockIdx.x * blockDim.x + threadIdx.x;
    if (id >= RT * ODP) return;
    int r = id / ODP, c = id % ODP;
    float v;
    if      (c < OD1)  v = H3[(size_t)r * ODP48 + c];
    else if (c == OD1) v = (r < R1) ? f1[(size_t)r * 3 + 2] : f2[(size_t)(r - R1) * 3 + 2];
    else               v = 0.f;
    X0[id] = v;
}

__global__ void k_deg(const float* __restrict__ Asub, int lda, int n,
                      float* __restrict__ dvec) {
    __shared__ float red[256];
    const int i = blockIdx.x, t = threadIdx.x;
    float cnt = 0.f;
    for (int j = t; j < n; j += 256)
        if (Asub[(size_t)i * lda + j] > 1e-15f) cnt += 1.f;
    float deg = blk_sum(cnt, red);
    if (t == 0) dvec[i] = rsqrtf(1.f + deg);
}

__global__ void k_ahat(const float* __restrict__ Asub, int lda,
                       const float* __restrict__ dvec, int n, float* __restrict__ Ah) {
    int id = blockIdx.x * blockDim.x + threadIdx.x;
    if (id >= n * n) return;
    int i = id / n, j = id % n;
    Ah[id] = dvec[i] * Asub[(size_t)i * lda + j] * dvec[j];
}

// theta = Wk Wk^T, padded to 36x48 with zeros
__global__ void k_theta(const float* __restrict__ Wk, float* __restrict__ Th) {
    int id = blockIdx.x * blockDim.x + threadIdx.x;
    if (id >= ODP * ODP48) return;
    int i = id / ODP48, j = id % ODP48;
    float d = 0.f;
    if (i < OD1 && j < OD1)
        for (int k = 0; k < 128; ++k) d += Wk[i * 128 + k] * Wk[j * 128 + k];
    Th[id] = d;
}

// xk stride 36: t<256 -> x1g(48), else H3(48) future rows; pad cols 0
__global__ void k_xk(const float* __restrict__ X1G, const float* __restrict__ H3,
                     float* __restrict__ XK) {
    int id = blockIdx.x * blockDim.x + threadIdx.x;
    if (id >= RT * ODP) return;
    int r = id / ODP, c = id % ODP;
    int b = r / NT, t = r % NT;
    float v = 0.f;
    if (c < OD1)
        v = (t < PASTN) ? X1G[((size_t)b * PASTN + t) * ODP48 + c]
                        : H3[((size_t)(R1 + b * FUT + (t - PASTN))) * ODP48 + c];
    XK[id] = v;
}

__global__ void k_q(const float* __restrict__ Z, const float* __restrict__ XK,
                    float* __restrict__ q) {
    int r = blockIdx.x * blockDim.x + threadIdx.x;
    if (r >= RT) return;
    float s = 0.f;
    for (int c = 0; c < OD1; ++c) s += Z[(size_t)r * ODP + c] * XK[(size_t)r * ODP + c];
    q[r] = s;
}

__global__ void k_At(const float* __restrict__ Aws, float* __restrict__ At) {
    __shared__ float red[256];
    const int f = blockIdx.x, p = threadIdx.x;
    float v = Aws[(size_t)(PASTN + f) * NT + p];
    float a = (v == 0.f) ? -__builtin_inff() : v;
    float mx = blk_max(a, red);
    float e = __expf(a - mx);
    float sm = blk_sum(e, red);
    At[f * PASTN + p] = e / sm;
}

__global__ void k_alpha(const float* __restrict__ Z, const float* __restrict__ XK,
                        const float* __restrict__ q, const float* __restrict__ At,
                        const float* __restrict__ X0, const float* __restrict__ smooth,
                        float* __restrict__ yh) {
    __shared__ float zf[OD1];
    __shared__ float red[256];
    const int f = blockIdx.x, b = blockIdx.y, p = threadIdx.x;
    const int rf = b * NT + PASTN + f;
    if (p < OD1) zf[p] = Z[(size_t)rf * ODP + p];
    __syncthreads();
    const int rp = b * NT + p;
    float cross = 0.f;
    for (int c = 0; c < OD1; ++c) cross += zf[c] * XK[(size_t)rp * ODP + c];
    float raw = q[rp] + q[rf] - 2.f * cross;
    float sg = 1.f / (1.f + __expf(-smooth[0]));
    float a = raw * (-0.5f / (sg * 0.01f));
    if (At[f * PASTN + p] < 0.005f) a = -__builtin_inff();
    float mx = blk_max(a, red);
    float e = __expf(a - mx);
    float den = blk_sum(e, red);
    float yv = X0[((size_t)b * PASTN + p) * ODP + OD1];
    float num = blk_sum(e * yv, red);
    if (p == 0) yh[b * FUT + f] = num / den;
}

__global__ void k_x2b(const float* __restrict__ H3, const float* __restrict__ yh,
                      float* __restrict__ X2B) {
    int id = blockIdx.x * blockDim.x + threadIdx.x;
    if (id >= R2 * ODP) return;
    int r = id / ODP, c = id % ODP;
    float v;
    if      (c < OD1)  v = H3[(size_t)(R1 + r) * ODP48 + c];
    else if (c == OD1) v = yh[r];
    else               v = 0.f;
    X2B[id] = v;
}

__global__ void k_t7(const float* __restrict__ T6, const float* __restrict__ g2W2,
                     float* __restrict__ T7) {
    int r = blockIdx.x * blockDim.x + threadIdx.x;
    if (r >= R2) return;
    float s = 0.f;
    for (int k = 0; k < 64; ++k) s += T6[(size_t)r * 64 + k] * g2W2[k];
    T7[r] = s;
}

__global__ void k_out(const float* __restrict__ AhF, const float* __restrict__ T7,
                      float* __restrict__ outp) {
    int id = blockIdx.x * blockDim.x + threadIdx.x;
    if (id >= SB * FUT) return;
    int b = id / FUT, i = id % FUT;
    float s = 0.f;
    for (int j = 0; j < FUT; ++j) s += AhF[i * FUT + j] * T7[b * FUT + j];
    outp[id] = s;
}

// xc (RT x 36), straight copy of padded x0full / x2b rows
__global__ void k_xc(const float* __restrict__ X0, const float* __restrict__ X2B,
                     float* __restrict__ XC) {
    int id = blockIdx.x * blockDim.x + threadIdx.x;
    if (id >= RT * ODP) return;
    int r = id / ODP, c = id % ODP;
    int b = r / NT, t = r % NT;
    XC[id] = (t < PASTN) ? X0[((size_t)b * PASTN + t) * ODP + c]
                         : X2B[((size_t)b * FUT + (t - PASTN)) * ODP + c];
}

__global__ void k_sq(const float* __restrict__ XC, float* __restrict__ sq) {
    int r = blockIdx.x * blockDim.x + threadIdx.x;
    if (r >= SB * NT) return;
    float s = 0.f;
    for (int k = 0; k < OD; ++k) { float v = XC[(size_t)r * ODP + k]; s += v * v; }
    sq[r] = s;
}

// cd[b] = sqrt(max(sq_m + sq_n - 2 * xc xc^T, 0)); batched WMMA Gram, K=36 padded
__global__ void k_cd(const float* __restrict__ XC, const float* __restrict__ sq,
                     float* __restrict__ cd) {
    __shared__ float bs[16 * ODP];
    const int b = blockIdx.y;
    const float* Xb = XC + (size_t)b * NT * ODP;
    const int ntn = NT >> 4;                       // 20
    const int bm = blockIdx.x / ntn, bn = blockIdx.x % ntn;
    const int n0 = bn << 4;
    for (int idx = threadIdx.x; idx < 16 * ODP; idx += 128) {
        int cc = idx / ODP, k = idx - cc * ODP;
        bs[idx] = Xb[(size_t)(n0 + cc) * ODP + k]; // contiguous row copy
    }
    __syncthreads();
    const int lane = threadIdx.x & 31;
    const int r = lane & 15, koff = (lane >> 4) << 1;
    const int m0 = (bm << 6) + ((threadIdx.x >> 5) << 4);
    const float* Arow = Xb + (size_t)(m0 + r) * ODP;
    __builtin_prefetch(Arow, 0, 3);
    v8f acc = wmma_loop(Arow, bs + r * ODP, ODP, koff);
    const int col = lane & 15, rhi = (lane >> 4) << 3;
    const float* sqb = sq + b * NT;
    for (int i = 0; i < 8; ++i) {
        int m = m0 + rhi + i, n = n0 + col;
        float d2 = sqb[m] + sqb[n] - 2.f * acc[i];
        cd[(size_t)b * NT * NT + (size_t)m * NT + n] = sqrtf(fmaxf(d2, 0.f));
    }
}

// ================= launch =================
extern "C" void kernel_launch(void* const* d_in, const int* in_sizes, int n_in,
                              void* d_out, int out_size, void* d_ws, size_t ws_size,
                              hipStream_t stream) {
    const float* Mg   = (const float*)d_in[0];
    const float* eh   = (const float*)d_in[1];
    const float* ed   = (const float*)d_in[2];
    const float* em   = (const float*)d_in[3];
    const float* pW1  = (const float*)d_in[4];
    const float* pb1  = (const float*)d_in[5];
    const float* pW2  = (const float*)d_in[6];
    const float* pb2  = (const float*)d_in[7];
    const float* pW3  = (const float*)d_in[8];
    const float* pb3  = (const float*)d_in[9];
    const float* g1W1 = (const float*)d_in[10];
    const float* g1W2 = (const float*)d_in[11];
    const float* Wk   = (const float*)d_in[12];
    const float* smo  = (const float*)d_in[13];
    const float* g2W1 = (const float*)d_in[14];
    const float* g2W2 = (const float*)d_in[15];
    const float* f1   = (const float*)d_in[16];
    const float* f2   = (const float*)d_in[17];
    const int*   c1   = (const int*)d_in[18];
    const int*   c2   = (const int*)d_in[19];

    float* ws   = (float*)d_ws;
    float* outp = (float*)d_out;
    float* cdp  = outp + SB * FUT;
    float* Aout = cdp + (size_t)SB * NT * NT;

    auto gb = [](int M, int N) { return (M / 64) * ((N + 15) / 16); };

    // 1. A
    k_rowA<<<NT, 256, 0, stream>>>(Mg, ws + oA, Aout);
    // weight padding
    k_padw<<<18, 256, 0, stream>>>(pW1, OD, 128, ws + oWp1, ODP, 128);
    k_padw<<<24, 256, 0, stream>>>(pW3, 128, OD1, ws + oWp3, 128, ODP48);
    k_padw<<<9,  256, 0, stream>>>(g1W1, OD, 64, ws + oG1W1, ODP, 64);
    k_padw<<<12, 256, 0, stream>>>(g1W2, 64, OD1, ws + oG1W2, 64, ODP48);
    k_padw<<<9,  256, 0, stream>>>(g2W1, OD, 64, ws + oG2W1, ODP, 64);
    k_theta<<<7, 256, 0, stream>>>(Wk, ws + oTh);
    // 2. features + MLP
    k_feat<<<(RT * ODP + 255) / 256, 256, 0, stream>>>(eh, ed, em, c1, f1, c2, f2, ws + oF);
    k_gemm<<<dim3(gb(RT, 128), 1), 128, 0, stream>>>(ws + oF, ODP, 0, ws + oWp1, 128, 0,
        pb1, ws + oP1, 128, 0, RT, 128, ODP, 1);
    k_gemm<<<dim3(gb(RT, 128), 1), 128, 0, stream>>>(ws + oP1, 128, 0, pW2, 128, 0,
        pb2, ws + oP2, 128, 0, RT, 128, 128, 1);
    k_gemm<<<dim3(gb(RT, OD1), 1), 128, 0, stream>>>(ws + oP2, 128, 0, ws + oWp3, ODP48, 0,
        pb3, ws + oH3, ODP48, 0, RT, OD1, 128, 0);
    k_x0build<<<(RT * ODP + 255) / 256, 256, 0, stream>>>(ws + oH3, f1, f2, ws + oX0);
    // 3. GCN on past
    k_deg<<<PASTN, 256, 0, stream>>>(ws + oA, NT, PASTN, ws + oDgP);
    k_ahat<<<256, 256, 0, stream>>>(ws + oA, NT, ws + oDgP, PASTN, ws + oAhP);
    k_gemm<<<dim3(gb(R1, 64), 1), 128, 0, stream>>>(ws + oX0, ODP, 0, ws + oG1W1, 64, 0,
        nullptr, ws + oT1, 64, 0, R1, 64, ODP, 0);
    k_gemm<<<dim3(gb(PASTN, 64), SB), 128, 0, stream>>>(ws + oAhP, PASTN, 0,
        ws + oT1, 64, (size_t)PASTN * 64, nullptr, ws + oT2, 64, (size_t)PASTN * 64,
        PASTN, 64, PASTN, 1);
    k_gemm<<<dim3(gb(R1, OD1), 1), 128, 0, stream>>>(ws + oT2, 64, 0, ws + oG1W2, ODP48, 0,
        nullptr, ws + oT3, ODP48, 0, R1, OD1, 64, 0);
    k_gemm<<<dim3(gb(PASTN, OD1), SB), 128, 0, stream>>>(ws + oAhP, PASTN, 0,
        ws + oT3, ODP48, (size_t)PASTN * ODP48, nullptr, ws + oX1G, ODP48,
        (size_t)PASTN * ODP48, PASTN, OD1, PASTN, 0);
    // 4. kernel attention
    k_xk<<<(RT * ODP + 255) / 256, 256, 0, stream>>>(ws + oX1G, ws + oH3, ws + oXK);
    k_gemm<<<dim3(gb(RT, OD1), 1), 128, 0, stream>>>(ws + oXK, ODP, 0, ws + oTh, ODP48, 0,
        nullptr, ws + oZ, ODP, 0, RT, OD1, ODP, 0);
    k_q<<<(RT + 255) / 256, 256, 0, stream>>>(ws + oZ, ws + oXK, ws + oQ);
    k_At<<<FUT, 256, 0, stream>>>(ws + oA, ws + oAt);
    k_alpha<<<dim3(FUT, SB), 256, 0, stream>>>(ws + oZ, ws + oXK, ws + oQ, ws + oAt,
                                               ws + oX0, smo, ws + oYh);
    // 5. GCN on future
    k_x2b<<<(R2 * ODP + 255) / 256, 256, 0, stream>>>(ws + oH3, ws + oYh, ws + oX2B);
    k_deg<<<FUT, 256, 0, stream>>>(ws + oA + (size_t)PASTN * NT + PASTN, NT, FUT, ws + oDgF);
    k_ahat<<<16, 256, 0, stream>>>(ws + oA + (size_t)PASTN * NT + PASTN, NT, ws + oDgF,
                                   FUT, ws + oAhF);
    k_gemm<<<dim3(gb(R2, 64), 1), 128, 0, stream>>>(ws + oX2B, ODP, 0, ws + oG2W1, 64, 0,
        nullptr, ws + oT5, 64, 0, R2, 64, ODP, 0);
    k_gemm<<<dim3(gb(FUT, 64), SB), 128, 0, stream>>>(ws + oAhF, FUT, 0,
        ws + oT5, 64, (size_t)FUT * 64, nullptr, ws + oT6, 64, (size_t)FUT * 64,
        FUT, 64, FUT, 1);
    k_t7<<<(R2 + 255) / 256, 256, 0, stream>>>(ws + oT6, g2W2, ws + oT7);
    k_out<<<(SB * FUT + 255) / 256, 256, 0, stream>>>(ws + oAhF, ws + oT7, outp);
    // 6. pairwise distances
    k_xc<<<(RT * ODP + 255) / 256, 256, 0, stream>>>(ws + oX0, ws + oX2B, ws + oXC);
    k_sq<<<(SB * NT + 255) / 256, 256, 0, stream>>>(ws + oXC, ws + oSq);
    k_cd<<<dim3((NT / 64) * (NT / 16), SB), 128, 0, stream>>>(ws + oXC, ws + oSq, cdp);
    (void)in_sizes; (void)n_in; (void)out_size; (void)ws_size;
}